// PagedAttention_74371653698021
// MI455X (gfx1250) — compile-verified
//
#include <hip/hip_runtime.h>
#include <math.h>

typedef float v2f __attribute__((ext_vector_type(2)));
typedef float v8f __attribute__((ext_vector_type(8)));
typedef unsigned int u32x4 __attribute__((ext_vector_type(4)));
typedef int i32x8 __attribute__((ext_vector_type(8)));
typedef int i32x4 __attribute__((ext_vector_type(4)));

#define NB     16          // B: sequences
#define HQ_    32          // query heads
#define HKV_   8           // kv heads
#define G_     4           // GQA group (HQ/HKV)
#define D_     128         // head dim
#define BS_    16          // cache block size
#define MAXB_  256         // blocks per sequence
#define MAXKV_ 4096
#define NSPLIT 16          // sequence partitions (flash-decoding splits)
#define PART   (MAXKV_ / NSPLIT)   // 256 keys per partition
#define TPP    (PART / BS_)        // 16 tiles per partition
#define NPART  NSPLIT              // 16 partial slots per (b,h,g)
#define KSTR   132         // LDS row stride (dwords): 128 + 4 pad (TDM pad reproduces this)
#define PSTR   20          // padded probs row stride (dwords)
#define SCALE_ 0.08838834764831845f   // 1/sqrt(128)
#define NEGF   (-1.0e30f)

#if defined(__HIP_DEVICE_COMPILE__) && __has_builtin(__builtin_amdgcn_tensor_load_to_lds) && __has_builtin(__builtin_amdgcn_s_wait_tensorcnt)
#define HAVE_TDM 1
#else
#define HAVE_TDM 0
#endif

#if HAVE_TDM
// Issue one TDM 2D tile load: 16 rows x 128 fp32, row stride HKV_*D_ elements in
// memory, padded to 132-dword rows in LDS; rows >= valid_rows OOB -> zero-filled.
__device__ __forceinline__ void tdm_load_tile(const float* gsrc, unsigned lds_byte_off,
                                              int valid_rows)
{
    const unsigned long long ga = (unsigned long long)(uintptr_t)gsrc;
    u32x4 g0;
    g0.x = 1u;                                           // count=1 (valid user D#)
    g0.y = lds_byte_off;                                 // lds_addr
    g0.z = (unsigned)(ga & 0xFFFFFFFFu);                 // global_addr[31:0]
    g0.w = (unsigned)((ga >> 32) & 0x1FFFFFFu)           // global_addr[56:32]
         | (2u << 30);                                   // type=2 ("image")
    i32x8 g1;
    g1[0] = (2 << 16)                                    // data_size = 4 bytes
          | (1 << 20)                                    // pad_enable
          | (6 << 22)                                    // pad_interval: 128 dwords
          | (3 << 25);                                   // pad_amount: 4 dwords
    g1[1] = (int)(128u << 16);                           // tensor_dim0 = 128 (row len)
    g1[2] = (int)((unsigned)valid_rows << 16);           // tensor_dim1 = valid rows
    g1[3] = (int)(128u << 16);                           // tile_dim0 = 128
    g1[4] = 16;                                          // tile_dim1 = 16 rows
    g1[5] = HKV_ * D_;                                   // tensor_dim0_stride = 1024 elems
    g1[6] = 0;
    g1[7] = 0;
    const i32x4 z4 = {0, 0, 0, 0};
    const i32x8 z8 = {0, 0, 0, 0, 0, 0, 0, 0};
    __builtin_amdgcn_tensor_load_to_lds(g0, g1, z4, z4, z8, 0);
}
#endif

// ---------------------------------------------------------------------------
// Kernel 1: one wave32 per (b, kv-head, split). TDM double-buffers K/V tiles
// into LDS; fp32 WMMA (v_wmma_f32_16x16x4_f32) does Q*K^T and P*V.
// ---------------------------------------------------------------------------
__global__ __launch_bounds__(32) void pa_partial_kernel(
    const float* __restrict__ q,  const float* __restrict__ kn, const float* __restrict__ vn,
    const float* __restrict__ kc, const float* __restrict__ vc,
    const int*   __restrict__ bt, const int*   __restrict__ slens,
    float* __restrict__ pout, float* __restrict__ pm, float* __restrict__ pl)
{
    const int bh    = blockIdx.x;          // 0..B*HKV-1
    const int b     = bh / HKV_;
    const int h     = bh % HKV_;
    const int split = blockIdx.y;          // 0..NSPLIT-1
    const int lane  = threadIdx.x & 31;
    const int hi    = lane >> 4;           // half-wave (K-dim split of fragments)
    const int ln    = lane & 15;           // M (A) / N (B,C) index

    __shared__ float qs[16 * KSTR];        // zero-padded Q tile (M=16)
    __shared__ float kbuf[2][16 * KSTR];   // double-buffered K tiles
    __shared__ float vbuf[2][16 * KSTR];   // double-buffered V tiles
    __shared__ float ps[16 * PSTR];        // probs C-layout -> A-layout bounce

    const int slen = slens[b];
    const int last = slen - 1;

    // ---- stage Q: 16 rows x 128 (rows >= G_ zero padded) ----
    for (int i = lane; i < 16 * 32; i += 32) {
        const int row = i >> 5;
        const int c4  = (i & 31) * 4;
        float4 val = make_float4(0.f, 0.f, 0.f, 0.f);
        if (row < G_)
            val = *(const float4*)(q + ((b * HQ_ + h * G_ + row) * D_ + c4));
        *(float4*)(qs + row * KSTR + c4) = val;
    }
    __syncthreads();

    const int part_base = split * PART;
    int ntiles = 0;
    if (slen > part_base) {
        int tt = (slen - part_base + BS_ - 1) / BS_;
        ntiles = tt < TPP ? tt : TPP;
    }

    v8f acc[8];                            // [dtile] C-layout accumulators
    #pragma unroll
    for (int dt = 0; dt < 8; ++dt) acc[dt] = (v8f){0,0,0,0,0,0,0,0};
    float m_run[8], l_run[8];
    #pragma unroll
    for (int r = 0; r < 8; ++r) { m_run[r] = NEGF; l_run[r] = 0.f; }

    const float* knew = kn + (b * HKV_ + h) * D_;
    const float* vnew = vn + (b * HKV_ + h) * D_;

#if HAVE_TDM
    // prime: issue K/V tensor loads for tile 0 into buffer 0
    if (ntiles > 0) {
        const int blk0 = bt[b * MAXB_ + split * TPP];
        int vr = slen - part_base; vr = vr > BS_ ? BS_ : vr;
        tdm_load_tile(kc + (blk0 * BS_ * HKV_ + h) * D_,
                      (unsigned)(uintptr_t)&kbuf[0][0], vr);
        tdm_load_tile(vc + (blk0 * BS_ * HKV_ + h) * D_,
                      (unsigned)(uintptr_t)&vbuf[0][0], vr);
    }
#endif

    for (int t = 0; t < ntiles; ++t) {
        const int buf      = t & 1;
        const int tile_pos = part_base + t * BS_;
        float* ksw = &kbuf[buf][0];
        float* vsw = &vbuf[buf][0];

#if HAVE_TDM
        // issue next tile's DMA into the other buffer, then wait for this tile
        if (t + 1 < ntiles) {
            const int nblk = bt[b * MAXB_ + split * TPP + t + 1];
            int vr = slen - (tile_pos + BS_); vr = vr > BS_ ? BS_ : vr;
            const int nb2 = (t + 1) & 1;
            tdm_load_tile(kc + (nblk * BS_ * HKV_ + h) * D_,
                          (unsigned)(uintptr_t)&kbuf[nb2][0], vr);
            tdm_load_tile(vc + (nblk * BS_ * HKV_ + h) * D_,
                          (unsigned)(uintptr_t)&vbuf[nb2][0], vr);
            __builtin_amdgcn_s_wait_tensorcnt(2);   // tile t's pair complete
        } else {
            __builtin_amdgcn_s_wait_tensorcnt(0);
        }
        // newest-token row substitution (caches are stale at that slot and we
        // must not mutate them in place)
        if (last >= tile_pos && last < tile_pos + BS_) {
            const int row = last - tile_pos;
            *(float4*)(ksw + row * KSTR + lane * 4) = *(const float4*)(knew + lane * 4);
            *(float4*)(vsw + row * KSTR + lane * 4) = *(const float4*)(vnew + lane * 4);
        }
#else
        // fallback: synchronous coalesced staging (one 512B row per load)
        const int blk   = bt[b * MAXB_ + split * TPP + t];
        const float* kb = kc + (blk * BS_ * HKV_ + h) * D_;
        const float* vb = vc + (blk * BS_ * HKV_ + h) * D_;
        if (t + 1 < ntiles) {
            const int nblk = bt[b * MAXB_ + split * TPP + t + 1];
            const float* kpf = kc + (nblk * BS_ * HKV_ + h) * D_;
            const float* vpf = vc + (nblk * BS_ * HKV_ + h) * D_;
            const int prow = lane >> 1, pcol = (lane & 1) * 64;
            __builtin_prefetch(kpf + prow * HKV_ * D_ + pcol, 0, 1);
            __builtin_prefetch(vpf + prow * HKV_ * D_ + pcol, 0, 1);
        }
        for (int row = 0; row < 16; ++row) {
            const int pos = tile_pos + row;                  // uniform
            float4 kval = make_float4(0.f, 0.f, 0.f, 0.f);
            float4 vval = kval;
            if (pos == last) {
                kval = *(const float4*)(knew + lane * 4);
                vval = *(const float4*)(vnew + lane * 4);
            } else if (pos < slen) {
                kval = *(const float4*)(kb + row * HKV_ * D_ + lane * 4);
                vval = *(const float4*)(vb + row * HKV_ * D_ + lane * 4);
            }
            *(float4*)(ksw + row * KSTR + lane * 4) = kval;
            *(float4*)(vsw + row * KSTR + lane * 4) = vval;
        }
#endif

        // ---- scores = Q * K^T : 32 chunks of v_wmma_f32_16x16x4_f32 ----
        // A[m][kk] = Q[m][4ch+kk], B[kk][n] = K[n][4ch+kk]
        v8f c = (v8f){0,0,0,0,0,0,0,0};
        #pragma unroll
        for (int ch = 0; ch < 32; ++ch) {
            const int dbase = ch * 4 + 2 * hi;
            v2f a  = *(const v2f*)(qs  + ln * KSTR + dbase);
            v2f kk = *(const v2f*)(ksw + ln * KSTR + dbase);
            c = __builtin_amdgcn_wmma_f32_16x16x4_f32(false, a, false, kk,
                                                      (short)0, c, false, false);
        }

        // ---- online softmax (C-layout: row r+8*hi, col ln) ----
        const bool colv = (tile_pos + ln) < slen;
        #pragma unroll
        for (int r = 0; r < 8; ++r) {
            float s = colv ? c[r] * SCALE_ : NEGF;
            float x = s;
            x = fmaxf(x, __shfl_xor(x, 1, 32));
            x = fmaxf(x, __shfl_xor(x, 2, 32));
            x = fmaxf(x, __shfl_xor(x, 4, 32));
            x = fmaxf(x, __shfl_xor(x, 8, 32));
            const float mnew = fmaxf(m_run[r], x);
            const float corr = __expf(m_run[r] - mnew);
            const float p    = __expf(s - mnew);
            float lsum = p;
            lsum += __shfl_xor(lsum, 1, 32);
            lsum += __shfl_xor(lsum, 2, 32);
            lsum += __shfl_xor(lsum, 4, 32);
            lsum += __shfl_xor(lsum, 8, 32);
            l_run[r] = l_run[r] * corr + lsum;
            m_run[r] = mnew;
            #pragma unroll
            for (int dt = 0; dt < 8; ++dt) acc[dt][r] *= corr;
            ps[(r + 8 * hi) * PSTR + ln] = p;
        }

        // ---- acc += P * V : 4 K-chunks x 8 N-tiles of WMMA ----
        #pragma unroll
        for (int kcn = 0; kcn < 4; ++kcn) {
            v2f pa = *(const v2f*)(ps + ln * PSTR + 4 * kcn + 2 * hi);
            #pragma unroll
            for (int dt = 0; dt < 8; ++dt) {
                v2f vvb;
                vvb.x = vsw[(4 * kcn + 2 * hi)     * KSTR + dt * 16 + ln];
                vvb.y = vsw[(4 * kcn + 2 * hi + 1) * KSTR + dt * 16 + ln];
                acc[dt] = __builtin_amdgcn_wmma_f32_16x16x4_f32(false, pa, false, vvb,
                                                                (short)0, acc[dt], false, false);
            }
        }
    }

    // ---- write this wave's partial: rows 0..G_-1 live in hi==0, VGPR r=g ----
    const int pslot = split;
    if (hi == 0) {
        #pragma unroll
        for (int g = 0; g < G_; ++g) {
            const int base = ((b * HKV_ + h) * G_ + g) * NPART + pslot;
            if (ln == 0) { pm[base] = m_run[g]; pl[base] = l_run[g]; }
            #pragma unroll
            for (int dt = 0; dt < 8; ++dt)
                pout[(size_t)base * D_ + dt * 16 + ln] = acc[dt][g];
        }
    }
}

// ---------------------------------------------------------------------------
// Kernel 2: combine NPART partials per output row (flash-decoding reduction).
// ---------------------------------------------------------------------------
__global__ __launch_bounds__(128) void pa_reduce_kernel(
    const float* __restrict__ pout, const float* __restrict__ pm,
    const float* __restrict__ pl,   float* __restrict__ out)
{
    const int idx = blockIdx.x;        // (b*HKV + h)*G + g == b*HQ + hq
    const int d   = threadIdx.x;       // 0..127
    __shared__ float sm[NPART], sl[NPART];
    if (d < NPART) {
        sm[d] = pm[idx * NPART + d];
        sl[d] = pl[idx * NPART + d];
    }
    __syncthreads();
    float M = NEGF;
    #pragma unroll
    for (int p = 0; p < NPART; ++p) M = fmaxf(M, sm[p]);
    float L = 0.f, o = 0.f;
    #pragma unroll
    for (int p = 0; p < NPART; ++p) {
        const float w = __expf(sm[p] - M);
        L += sl[p] * w;
        o += w * pout[((size_t)idx * NPART + p) * D_ + d];
    }
    out[(size_t)idx * D_ + d] = o / L;
}

// ---------------------------------------------------------------------------
extern "C" void kernel_launch(void* const* d_in, const int* in_sizes, int n_in,
                              void* d_out, int out_size, void* d_ws, size_t ws_size,
                              hipStream_t stream)
{
    (void)in_sizes; (void)n_in; (void)out_size; (void)ws_size;
    const float* q  = (const float*)d_in[0];
    const float* k  = (const float*)d_in[1];
    const float* v  = (const float*)d_in[2];
    const float* kc = (const float*)d_in[3];
    const float* vc = (const float*)d_in[4];
    // d_in[5] = slot_mapping (unused: newest token substituted at gather time)
    const int* bt   = (const int*)d_in[6];
    const int* sl   = (const int*)d_in[7];
    float* out = (float*)d_out;

    float* pout = (float*)d_ws;                                      // [B*HKV*G*NPART*D]
    float* pm   = pout + (size_t)NB * HKV_ * G_ * NPART * D_;        // [B*HKV*G*NPART]
    float* pl   = pm   + (size_t)NB * HKV_ * G_ * NPART;             // [B*HKV*G*NPART]

    dim3 g1(NB * HKV_, NSPLIT);
    pa_partial_kernel<<<g1, 32, 0, stream>>>(q, k, v, kc, vc, bt, sl, pout, pm, pl);
    pa_reduce_kernel<<<NB * HQ_, D_, 0, stream>>>(pout, pm, pl, out);
}